// MKMMDLoss_13082470384135
// MI455X (gfx1250) — compile-verified
//
#include <hip/hip_runtime.h>
#include <hip/hip_bf16.h>
#include <math.h>

typedef __attribute__((ext_vector_type(2))) float v2f;
typedef __attribute__((ext_vector_type(8))) float v8f;

#define NROWS 2048
#define DIMF  512
#define KNUM  5

// ws layout in floats:
//   [0,2048)          ns  = ||Xs_i||^2
//   [2048,4096)       nt  = ||Xt_i||^2
//   [4096,14336)      per-WG partial sums (2048 WGs x 5 gammas)
//   [14336,14341)     eta_k
//   [14344,19464)     h[5][1024]
#define WS_NS   0
#define WS_NT   2048
#define WS_PART 4096
#define WS_ETA  14336
#define WS_H    14344

// 1/(2*gamma^2) for gamma = 2, 1, 0.5, 0.25, 0.125
__device__ __constant__ float c_ig[KNUM] = {0.125f, 0.5f, 2.0f, 8.0f, 32.0f};

__device__ __forceinline__ float wave_reduce(float v) {
#pragma unroll
  for (int off = 16; off > 0; off >>= 1) v += __shfl_down(v, off, 32);
  return v;
}

// ---------------- row norms: one wave per row (Xs rows then Xt rows) --------
__global__ __launch_bounds__(256) void norms_kernel(const float* __restrict__ Xs,
                                                    const float* __restrict__ Xt,
                                                    float* __restrict__ ws) {
  int lane = threadIdx.x & 31, wave = threadIdx.x >> 5;
  int w = blockIdx.x * 8 + wave;                 // 0..4095
  const float* row = (w < NROWS) ? (Xs + w * DIMF) : (Xt + (w - NROWS) * DIMF);
  float s = 0.f;
  for (int e = lane; e < DIMF; e += 32) { float x = row[e]; s += x * x; }
  s = wave_reduce(s);
  if (lane == 0) ws[w] = s;                      // ns then nt contiguous
}

// ---------------- fused Gram (WMMA f32) + RBF exp reduction -----------------
// Grid: (2048/32, 2048/64) WGs of 256 threads. Each wave: one 16x16 C tile.
__global__ __launch_bounds__(256) void gram_exp_kernel(const float* __restrict__ Xs,
                                                       const float* __restrict__ Xt,
                                                       float* __restrict__ ws) {
  __shared__ float red[256];
  const float* ns = ws + WS_NS;
  const float* nt = ws + WS_NT;
  float* partial  = ws + WS_PART;

  const int lane = threadIdx.x & 31;
  const int wave = threadIdx.x >> 5;
  const int rowI = blockIdx.y * 64 + (wave & 3) * 16;   // M tile base
  const int colJ = blockIdx.x * 32 + (wave >> 2) * 16;  // N tile base
  const int m16  = lane & 15;
  const int kb   = (lane >> 4);  // half-wave K offset in float2 units (K +0 / +2)

  // A fragment: lane holds {X[rowI+m16][k+2*kb], X[rowI+m16][k+2*kb+1]}
  // B fragment for X*Y^T: identical gather on rows of Y (col index = m16)
  const v2f* as = (const v2f*)(Xs + (rowI + m16) * DIMF);
  const v2f* at = (const v2f*)(Xt + (rowI + m16) * DIMF);
  const v2f* bs = (const v2f*)(Xs + (colJ + m16) * DIMF);
  const v2f* bt = (const v2f*)(Xt + (colJ + m16) * DIMF);

  v8f css = {}, ctt = {}, cst = {};
#pragma unroll 4
  for (int k2 = 0; k2 < DIMF / 2; k2 += 2) {   // K advances by 4 floats
    v2f a_s = as[k2 + kb];
    v2f a_t = at[k2 + kb];
    v2f b_s = bs[k2 + kb];
    v2f b_t = bt[k2 + kb];
    css = __builtin_amdgcn_wmma_f32_16x16x4_f32(false, a_s, false, b_s,
                                                (short)0, css, false, false);
    ctt = __builtin_amdgcn_wmma_f32_16x16x4_f32(false, a_t, false, b_t,
                                                (short)0, ctt, false, false);
    cst = __builtin_amdgcn_wmma_f32_16x16x4_f32(false, a_s, false, b_t,
                                                (short)0, cst, false, false);
  }

  // C layout: VGPR r -> M = r + 8*(lane>=16), N = lane&15
  const int j = colJ + m16;
  const float nsj = ns[j], ntj = nt[j];
  const int ibase = rowI + ((lane >> 4) << 3);
  float dss[8], dtt[8], dst[8];
#pragma unroll
  for (int r = 0; r < 8; ++r) {
    int i = ibase + r;
    float nsi = ns[i], nti = nt[i];
    dss[r] = nsi + nsj - 2.f * css[r];
    dtt[r] = nti + ntj - 2.f * ctt[r];
    dst[r] = nsi + ntj - 2.f * cst[r];
  }

  const int wg = blockIdx.y * gridDim.x + blockIdx.x;
  for (int k = 0; k < KNUM; ++k) {
    float ig = c_ig[k];
    float s = 0.f;
#pragma unroll
    for (int r = 0; r < 8; ++r)
      s += __expf(-dss[r] * ig) + __expf(-dtt[r] * ig) - 2.f * __expf(-dst[r] * ig);
    red[threadIdx.x] = s;
    __syncthreads();
    for (int off = 128; off > 0; off >>= 1) {
      if (threadIdx.x < off) red[threadIdx.x] += red[threadIdx.x + off];
      __syncthreads();
    }
    if (threadIdx.x == 0) partial[wg * KNUM + k] = red[0];
    __syncthreads();
  }
}

// ---------------- deterministic reduction of per-WG partials -> eta ---------
__global__ __launch_bounds__(256) void eta_kernel(float* __restrict__ ws) {
  __shared__ float red[256];
  const float* partial = ws + WS_PART;
  float* eta = ws + WS_ETA;
  for (int k = 0; k < KNUM; ++k) {
    float s = 0.f;
    for (int idx = threadIdx.x; idx < 2048; idx += 256) s += partial[idx * KNUM + k];
    red[threadIdx.x] = s;
    __syncthreads();
    for (int off = 128; off > 0; off >>= 1) {
      if (threadIdx.x < off) red[threadIdx.x] += red[threadIdx.x + off];
      __syncthreads();
    }
    if (threadIdx.x == 0) eta[k] = red[0] / 4194304.f;  // n*n
    __syncthreads();
  }
}

// ---------------- h[k][j] = g[k, 2j, 2j+1]: one wave per j ------------------
__global__ __launch_bounds__(256) void h_kernel(const float* __restrict__ Xs,
                                                const float* __restrict__ Xt,
                                                float* __restrict__ ws) {
  const float* ns = ws + WS_NS;
  const float* nt = ws + WS_NT;
  float* h = ws + WS_H;
  int lane = threadIdx.x & 31, wave = threadIdx.x >> 5;
  int j = blockIdx.x * 8 + wave;                 // 0..1023
  const float* xs0 = Xs + (2 * j) * DIMF;
  const float* xs1 = Xs + (2 * j + 1) * DIMF;
  const float* xt0 = Xt + (2 * j) * DIMF;
  const float* xt1 = Xt + (2 * j + 1) * DIMF;
  float pss = 0.f, ptt = 0.f, pst = 0.f, pts = 0.f;
  for (int e = lane; e < DIMF; e += 32) {
    float a = xs0[e], b = xs1[e], c = xt0[e], d = xt1[e];
    pss += a * b; ptt += c * d; pst += a * d; pts += c * b;
  }
  pss = wave_reduce(pss); ptt = wave_reduce(ptt);
  pst = wave_reduce(pst); pts = wave_reduce(pts);
  if (lane == 0) {
    float dss = ns[2 * j] + ns[2 * j + 1] - 2.f * pss;
    float dtt = nt[2 * j] + nt[2 * j + 1] - 2.f * ptt;
    float dst = ns[2 * j] + nt[2 * j + 1] - 2.f * pst;
    float dts = nt[2 * j] + ns[2 * j + 1] - 2.f * pts;
    for (int k = 0; k < KNUM; ++k) {
      float ig = c_ig[k];
      h[k * 1024 + j] = __expf(-dss * ig) + __expf(-dtt * ig)
                      - __expf(-dst * ig) - __expf(-dts * ig);
    }
  }
}

// ---------------- Q/p build + 31-mask simplex QP + final dot ----------------
__global__ __launch_bounds__(256) void solve_kernel(float* __restrict__ ws,
                                                    float* __restrict__ out) {
  __shared__ float part[256][20];
  const float* h   = ws + WS_H;
  const float* eta = ws + WS_ETA;
  int tid = threadIdx.x;

  float acc[20];
#pragma unroll
  for (int q = 0; q < 20; ++q) acc[q] = 0.f;
  for (int m = tid; m < 512; m += 256) {
    float h4[KNUM], hs[KNUM];
    for (int k = 0; k < KNUM; ++k) {
      float a = h[k * 1024 + 2 * m], b = h[k * 1024 + 2 * m + 1];
      h4[k] = a - b; hs[k] = a + b;
    }
    int q = 0;
    for (int k1 = 0; k1 < KNUM; ++k1)
      for (int k2 = k1; k2 < KNUM; ++k2) acc[q++] += h4[k1] * h4[k2];
    for (int k = 0; k < KNUM; ++k) acc[15 + k] += hs[k];
  }
#pragma unroll
  for (int q = 0; q < 20; ++q) part[tid][q] = acc[q];
  __syncthreads();

  if (tid == 0) {
    float tot[20];
    for (int q = 0; q < 20; ++q) {
      float s = 0.f;
      for (int t = 0; t < 256; ++t) s += part[t][q];
      tot[q] = s;
    }
    float qpair[KNUM][KNUM];
    int q = 0;
    for (int k1 = 0; k1 < KNUM; ++k1)
      for (int k2 = k1; k2 < KNUM; ++k2) { qpair[k1][k2] = tot[q]; qpair[k2][k1] = tot[q]; ++q; }
    float Q[KNUM][KNUM], p[KNUM];
    for (int k1 = 0; k1 < KNUM; ++k1)
      for (int k2 = 0; k2 < KNUM; ++k2) {
        float Qp = (4.f / 2048.f) * (qpair[k1][k2] + ((k1 == k2) ? qpair[k1][k1] : 0.f));
        Q[k1][k2] = 2.f * Qp + ((k1 == k2) ? 1e-5f : 0.f);
      }
    for (int k = 0; k < KNUM; ++k) p[k] = -(2.f * tot[15 + k] / 2048.f);

    float best = INFINITY, bbeta[KNUM] = {0.f, 0.f, 0.f, 0.f, 0.f};
    for (int bits = 1; bits < 32; ++bits) {
      float mk[KNUM];
      for (int k = 0; k < KNUM; ++k) mk[k] = (float)((bits >> k) & 1);
      float M[6][7];
      for (int r = 0; r < 6; ++r)
        for (int c = 0; c < 7; ++c) M[r][c] = 0.f;
      for (int r = 0; r < KNUM; ++r) {
        for (int c = 0; c < KNUM; ++c) M[r][c] = mk[r] * Q[r][c] * mk[c];
        M[r][r] += 1.f - mk[r];
        M[r][5] = mk[r]; M[5][r] = mk[r];
        M[r][6] = -mk[r] * p[r];
      }
      M[5][6] = 1.f;
      bool ok = true;
      for (int c = 0; c < 6; ++c) {                 // Gauss-Jordan w/ pivoting
        int piv = c; float pv = fabsf(M[c][c]);
        for (int r = c + 1; r < 6; ++r)
          if (fabsf(M[r][c]) > pv) { pv = fabsf(M[r][c]); piv = r; }
        if (pv < 1e-12f) { ok = false; break; }
        if (piv != c)
          for (int cc = c; cc < 7; ++cc) { float t = M[c][cc]; M[c][cc] = M[piv][cc]; M[piv][cc] = t; }
        float inv = 1.f / M[c][c];
        for (int r = 0; r < 6; ++r) if (r != c) {
          float f = M[r][c] * inv;
          for (int cc = c; cc < 7; ++cc) M[r][cc] -= f * M[c][cc];
        }
      }
      if (!ok) continue;
      float beta[KNUM]; bool feas = true;
      for (int k = 0; k < KNUM; ++k) {
        beta[k] = (M[k][6] / M[k][k]) * mk[k];
        if (beta[k] < -1e-7f) feas = false;
      }
      if (!feas) continue;
      float obj = 0.f;
      for (int r = 0; r < KNUM; ++r) {
        float t = 0.f;
        for (int c = 0; c < KNUM; ++c) t += Q[r][c] * beta[c];
        obj += 0.5f * beta[r] * t + p[r] * beta[r];
      }
      if (obj < best) { best = obj; for (int k = 0; k < KNUM; ++k) bbeta[k] = beta[k]; }
    }
    float res = 0.f;
    for (int k = 0; k < KNUM; ++k) res += eta[k] * bbeta[k];
    out[0] = res;   // SIGN = 1.0
  }
}

extern "C" void kernel_launch(void* const* d_in, const int* in_sizes, int n_in,
                              void* d_out, int out_size, void* d_ws, size_t ws_size,
                              hipStream_t stream) {
  const float* Xs = (const float*)d_in[0];
  const float* Xt = (const float*)d_in[1];
  float* ws  = (float*)d_ws;
  float* out = (float*)d_out;

  norms_kernel<<<512, 256, 0, stream>>>(Xs, Xt, ws);
  dim3 grid(NROWS / 32, NROWS / 64);   // 64 x 32 = 2048 workgroups
  gram_exp_kernel<<<grid, 256, 0, stream>>>(Xs, Xt, ws);
  eta_kernel<<<1, 256, 0, stream>>>(ws);
  h_kernel<<<128, 256, 0, stream>>>(Xs, Xt, ws);
  solve_kernel<<<1, 256, 0, stream>>>(ws, out);
}